// HONU_80865644249720
// MI455X (gfx1250) — compile-verified
//
#include <hip/hip_runtime.h>
#include <hip/hip_bf16.h>

// ---------------------------------------------------------------------------
// HONU order-3, IN_FEATURES=64 (+bias), BATCH=2048.
// out[b] = sum_{i<=j<=k} w[(i,j,k)] * xb[b,i]*xb[b,j]*xb[b,k],  xb=[1,x].
//
// Factorization: p indexes pairs (j<=k), NPAIR = 65*66/2 = 2145.
//   W2[p, i] = w[(i, j_p, k_p)] for i <= j_p, else 0   (scattered from weights)
//   Y[b, p]  = sum_i xb[b,i] * W2[p,i]                 (fp32 WMMA GEMM, K=65)
//   out[b]   = sum_p Y[b,p] * xb[b,j_p] * xb[b,k_p]    (fused epilogue)
// ---------------------------------------------------------------------------

#define NFEAT   65           // 1 + 64
#define KPAD    68           // 17 WMMA K-steps of 4
#define KSTEPS  17
#define NPAIR   2145
#define NPAD    2160         // 135 N-tiles of 16
#define NTILES  135
#define BATCH   2048
#define W2_FLOATS ((KPAD / 2) * NPAD * 2)   // 34 * 2160 * 2 = 146880 floats
#define W2_BYTES  (W2_FLOATS * 4)           // 587,520 bytes

typedef __attribute__((ext_vector_type(2))) float v2f;
typedef __attribute__((ext_vector_type(8))) float v8f;

// W2 storage: K-major with K-pairs interleaved so one b64 load fetches (k, k+1):
//   element (k, p) lives at W2[(k>>1)*(NPAD*2) + p*2 + (k&1)]

__global__ void honu_zero_w2(float* __restrict__ w2) {
    int i = blockIdx.x * blockDim.x + threadIdx.x;
    if (i < W2_FLOATS) w2[i] = 0.0f;
}

__global__ void honu_scatter_w(const int* __restrict__ comb,
                               const float* __restrict__ w,
                               float* __restrict__ w2, int nw) {
    int m = blockIdx.x * blockDim.x + threadIdx.x;
    if (m >= nw) return;
    int i = comb[3 * m + 0];
    int j = comb[3 * m + 1];
    int k = comb[3 * m + 2];
    int p = j * NFEAT - (j * (j - 1)) / 2 + (k - j);   // pair index, j<=k
    w2[(size_t)(i >> 1) * (NPAD * 2) + p * 2 + (i & 1)] = w[m];
}

__global__ void honu_build_pairs(int* __restrict__ pairJK) {
    int p = blockIdx.x * blockDim.x + threadIdx.x;
    if (p >= NPAD) return;
    int j = 0, k = 0;
    if (p < NPAIR) {
        int rem = p;
        for (int t = 0; t < NFEAT; ++t) {
            int cnt = NFEAT - t;            // pairs starting with j = t
            if (rem < cnt) { j = t; k = t + rem; break; }
            rem -= cnt;
        }
    }
    pairJK[p] = (j << 16) | k;              // padding -> (0,0): harmless, Y==0 there
}

__global__ __launch_bounds__(128)
void honu_main(const float* __restrict__ x,
               const float* __restrict__ w2,
               const int* __restrict__ pairJK,
               float* __restrict__ out) {
    __shared__ float xb[4][16][KPAD];       // per-wave 16-row xb tile (17,408 B)

    const int lane = threadIdx.x & 31;
    const int wave = threadIdx.x >> 5;
    const int tile = blockIdx.x * 4 + wave; // 0..127
    const int rowbase = tile * 16;

    // Build xb tile: col 0 = 1.0 (bias), cols 1..64 = x, cols 65..67 = 0 (K pad).
    // 16*68 = 1088 = 34*32 -> perfectly uniform loop, no divergence.
    for (int e = lane; e < 16 * KPAD; e += 32) {
        int r = e / KPAD;
        int c = e % KPAD;
        float v;
        if (c == 0)            v = 1.0f;
        else if (c < NFEAT)    v = x[(size_t)(rowbase + r) * 64 + (c - 1)];
        else                   v = 0.0f;
        xb[wave][r][c] = v;
    }
    __syncthreads();

    // A fragments for all 17 K-steps (32-bit 16x4 A layout):
    //   lanes 0-15: M=lane, K = base+{0,1}; lanes 16-31: M=lane-16, K = base+{2,3}
    const int m    = lane & 15;
    const int koff = (lane >> 4) << 1;      // 0 or 2
    const int mh   = (lane >> 4) << 3;      // C/D row offset: 0 or 8
    v2f a[KSTEPS];
#pragma unroll
    for (int kk = 0; kk < KSTEPS; ++kk) {
        a[kk].x = xb[wave][m][kk * 4 + koff + 0];
        a[kk].y = xb[wave][m][kk * 4 + koff + 1];
    }

    float acc[8];
#pragma unroll
    for (int r = 0; r < 8; ++r) acc[r] = 0.0f;

    for (int nt = 0; nt < NTILES; ++nt) {
        const int n = nt * 16 + m;          // this lane's pair column (B layout N=lane&15)
        v8f c = {};
#pragma unroll
        for (int kk = 0; kk < KSTEPS; ++kk) {
            const int k0 = kk * 4 + koff;   // even -> (k0, k0+1) packed adjacently
            const v2f b = *(const v2f*)(w2 + (size_t)(k0 >> 1) * (NPAD * 2) + n * 2);
            c = __builtin_amdgcn_wmma_f32_16x16x4_f32(
                    /*neg_a=*/false, a[kk], /*neg_b=*/false, b,
                    /*c_mod=*/(short)0, c, /*reuse_a=*/false, /*reuse_b=*/false);
        }
        // Epilogue: out contribution Y[row,n] * xb[row,j_n] * xb[row,k_n].
        // C/D layout: VGPR r holds (M = r + mh, N = lane&15) -> one (j,k) per lane.
        const int jk = pairJK[n];
        const int j  = jk >> 16;
        const int k  = jk & 0xFFFF;
#pragma unroll
        for (int r = 0; r < 8; ++r) {
            const int row = r + mh;
            acc[r] += c[r] * xb[wave][row][j] * xb[wave][row][k];
        }
    }

    // Reduce over the 16 lanes of each half (N was striped across lane&15).
#pragma unroll
    for (int r = 0; r < 8; ++r) {
        float v = acc[r];
        v += __shfl_xor(v, 1, 32);
        v += __shfl_xor(v, 2, 32);
        v += __shfl_xor(v, 4, 32);
        v += __shfl_xor(v, 8, 32);
        acc[r] = v;
    }
    if ((lane & 15) == 0) {
#pragma unroll
        for (int r = 0; r < 8; ++r)
            out[rowbase + mh + r] = acc[r];
    }
}

extern "C" void kernel_launch(void* const* d_in, const int* in_sizes, int n_in,
                              void* d_out, int out_size, void* d_ws, size_t ws_size,
                              hipStream_t stream) {
    const float* x       = (const float*)d_in[0];   // (2048, 64) f32
    const float* weights = (const float*)d_in[1];   // (47905,)   f32
    const int*   comb    = (const int*)d_in[2];     // (47905, 3) i32
    float*       out     = (float*)d_out;           // (2048,)    f32

    float* w2     = (float*)d_ws;                           // 587,520 B
    int*   pairJK = (int*)((char*)d_ws + W2_BYTES);         // 8,640 B

    const int nw = in_sizes[1];

    honu_zero_w2   <<<(W2_FLOATS + 255) / 256, 256, 0, stream>>>(w2);
    honu_build_pairs<<<(NPAD     + 255) / 256, 256, 0, stream>>>(pairJK);
    honu_scatter_w <<<(nw        + 255) / 256, 256, 0, stream>>>(comb, weights, w2, nw);

    // 128 waves total: 32 blocks x 4 waves, each wave owns a 16-row batch tile.
    honu_main<<<32, 128, 0, stream>>>(x, w2, pairJK, out);
    (void)n_in; (void)out_size; (void)ws_size;
}